// GAT_51762945851571
// MI455X (gfx1250) — compile-verified
//
#include <hip/hip_runtime.h>
#include <hip/hip_bf16.h>

#define IN_F  128
#define OUT_F 64
#define NEG_SLOPE 0.2f

typedef __attribute__((ext_vector_type(2))) float v2f;
typedef __attribute__((ext_vector_type(8))) float v8f;

// ---- orderable-uint mapping for float atomic max ----
__device__ __forceinline__ unsigned f2ord(float f) {
    unsigned u = __float_as_uint(f);
    return (u & 0x80000000u) ? ~u : (u | 0x80000000u);
}
__device__ __forceinline__ float ord2f(unsigned u) {
    unsigned v = (u & 0x80000000u) ? (u & 0x7FFFFFFFu) : ~u;
    return __uint_as_float(v);
}

// ---------------- K0: out = bias (broadcast), zero segment buffers ----------------
__global__ __launch_bounds__(256) void gat_init(float* __restrict__ out,
                                                const float* __restrict__ bias,
                                                unsigned* __restrict__ smax,
                                                float* __restrict__ ssum, int N) {
    int idx = blockIdx.x * 256 + threadIdx.x;
    if (idx < N * OUT_F) out[idx] = bias[idx & (OUT_F - 1)];
    if (idx < N) { smax[idx] = 0u; ssum[idx] = 0.0f; }
}

// ---------------- K1: h = x*W via V_WMMA_F32_16X16X4_F32; fused a_s/a_d ----------------
// One wave per 16-row tile; 8 waves (256 threads) per block -> 128 rows/block.
// LDS holds W re-staged K-pair interleaved: Ws[(k>>1)*128 + n*2 + (k&1)]
// so each lane's B fragment (two consecutive K for its column n) is one ds_load_b64.
__global__ __launch_bounds__(256) void gat_gemm(const float* __restrict__ x,
                                                const float* __restrict__ W,
                                                const float* __restrict__ att_src,
                                                const float* __restrict__ att_dst,
                                                float* __restrict__ h,
                                                float* __restrict__ a_s,
                                                float* __restrict__ a_d, int N) {
    __shared__ float Ws[IN_F * OUT_F];              // 32 KB of the 320 KB/WGP LDS
    for (int i = threadIdx.x; i < IN_F * OUT_F; i += 256) {
        int k = i >> 6;                             // W is [128][64] row-major
        int n = i & 63;
        Ws[(k >> 1) * 128 + n * 2 + (k & 1)] = W[i];
    }
    __syncthreads();

    const int lane = threadIdx.x & 31;
    const int wave = threadIdx.x >> 5;
    const int m    = lane & 15;                     // row (A) / column (B,C) within tile
    const int hi   = lane >> 4;                     // half-wave -> K-pair select / +8 rows in C
    const int row0 = (blockIdx.x * 8 + wave) * 16;

    const int arow = row0 + m;
    const int crow = arow < N ? arow : 0;           // clamp: loads always in-bounds,
                                                    // tail rows are store-guarded below
    const float2* xrow2 = (const float2*)(x + (size_t)crow * IN_F);

    v8f c0 = {}, c1 = {}, c2 = {}, c3 = {};

    for (int k0 = 0; k0 < IN_F; k0 += 4) {
        const int kp = (k0 >> 1) + hi;              // K-pair index for this half-wave
        float2 af = xrow2[kp];                      // global_load_b64, unconditional
        v2f a = { af.x, af.y };
        const float2* wrow = (const float2*)&Ws[kp * 128];
        float2 f0 = wrow[m];                        // ds_load_b64, conflict-free
        float2 f1 = wrow[m + 16];
        float2 f2 = wrow[m + 32];
        float2 f3 = wrow[m + 48];
        v2f b0 = { f0.x, f0.y };
        v2f b1 = { f1.x, f1.y };
        v2f b2 = { f2.x, f2.y };
        v2f b3 = { f3.x, f3.y };
        c0 = __builtin_amdgcn_wmma_f32_16x16x4_f32(false, a, false, b0, (short)0, c0, false, false);
        c1 = __builtin_amdgcn_wmma_f32_16x16x4_f32(false, a, false, b1, (short)0, c1, false, false);
        c2 = __builtin_amdgcn_wmma_f32_16x16x4_f32(false, a, false, b2, (short)0, c2, false, false);
        c3 = __builtin_amdgcn_wmma_f32_16x16x4_f32(false, a, false, b3, (short)0, c3, false, false);
    }

    // att vectors indexed by this lane's output column in each 16-wide N tile
    const float as0 = att_src[m],      as1 = att_src[16 + m];
    const float as2 = att_src[32 + m], as3 = att_src[48 + m];
    const float ad0 = att_dst[m],      ad1 = att_dst[16 + m];
    const float ad2 = att_dst[32 + m], ad3 = att_dst[48 + m];

#pragma unroll
    for (int r = 0; r < 8; ++r) {
        const int row = row0 + r + 8 * hi;          // C layout: VGPR r -> M = r (+8 upper half-wave)
        const float e0 = c0[r], e1 = c1[r], e2 = c2[r], e3 = c3[r];
        if (row < N) {
            float* hr = h + (size_t)row * OUT_F;
            hr[m]      = e0;
            hr[16 + m] = e1;
            hr[32 + m] = e2;
            hr[48 + m] = e3;
        }
        float ps = e0 * as0 + e1 * as1 + e2 * as2 + e3 * as3;
        float pd = e0 * ad0 + e1 * ad1 + e2 * ad2 + e3 * ad3;
#pragma unroll
        for (int off = 8; off >= 1; off >>= 1) {    // butterfly within each 16-lane N group
            ps += __shfl_xor(ps, off, 32);
            pd += __shfl_xor(pd, off, 32);
        }
        if (m == 0 && row < N) { a_s[row] = ps; a_d[row] = pd; }
    }
}

// ---------------- K2: per-edge logits, leaky relu, atomic segment max ----------------
__global__ __launch_bounds__(256) void gat_logits(const int* __restrict__ ei,
                                                  const float* __restrict__ a_s,
                                                  const float* __restrict__ a_d,
                                                  float* __restrict__ e,
                                                  unsigned* __restrict__ smax,
                                                  int E, int T) {
    int idx = blockIdx.x * 256 + threadIdx.x;
    if (idx >= T) return;
    int s, d;
    if (idx < E) { s = ei[idx]; d = ei[E + idx]; }
    else         { s = d = idx - E; }               // virtual self loops
    float v = a_s[s] + a_d[d];
    v = v > 0.0f ? v : NEG_SLOPE * v;
    e[idx] = v;
    atomicMax(&smax[d], f2ord(v));
}

// ---------------- K3: exp(e - max) + atomic segment sum ----------------
__global__ __launch_bounds__(256) void gat_exp(const int* __restrict__ ei,
                                               float* __restrict__ e,
                                               const unsigned* __restrict__ smax,
                                               float* __restrict__ ssum,
                                               int E, int T) {
    int idx = blockIdx.x * 256 + threadIdx.x;
    if (idx >= T) return;
    int d = (idx < E) ? ei[E + idx] : (idx - E);
    float w = __expf(e[idx] - ord2f(smax[d]));
    e[idx] = w;
    atomicAdd(&ssum[d], w);
}

// ---------------- K4: weighted scatter-add, one wave per edge, 2 cols/lane ----------------
__global__ __launch_bounds__(256) void gat_aggregate(const int* __restrict__ ei,
                                                     const float* __restrict__ h,
                                                     const float* __restrict__ e,
                                                     const float* __restrict__ ssum,
                                                     float* __restrict__ out,
                                                     int E, int T) {
    int edge = blockIdx.x * 8 + (threadIdx.x >> 5);
    if (edge >= T) return;
    int lane = threadIdx.x & 31;
    int s, d;
    if (edge < E) { s = ei[edge]; d = ei[E + edge]; }
    else          { s = d = edge - E; }
    float alpha = e[edge] / ssum[d];
    const float2* hv = (const float2*)(h + (size_t)s * OUT_F);
    float2 v = hv[lane];
    float* od = out + (size_t)d * OUT_F + 2 * lane;
    atomicAdd(od,     alpha * v.x);
    atomicAdd(od + 1, alpha * v.y);
}

extern "C" void kernel_launch(void* const* d_in, const int* in_sizes, int n_in,
                              void* d_out, int out_size, void* d_ws, size_t ws_size,
                              hipStream_t stream) {
    const float* x       = (const float*)d_in[0];
    const int*   ei      = (const int*)d_in[1];    // [2, E] flattened: row0 = src, row1 = dst
    const float* W       = (const float*)d_in[2];
    const float* att_src = (const float*)d_in[3];
    const float* att_dst = (const float*)d_in[4];
    const float* bias    = (const float*)d_in[5];

    const int N = in_sizes[0] / IN_F;
    const int E = in_sizes[1] / 2;
    const int T = E + N;                            // edges + self loops

    float* out = (float*)d_out;

    // workspace layout (floats): h[N*64] | a_s[N] | a_d[N] | e[T] | ssum[N] | smax[N]
    float*    h    = (float*)d_ws;
    float*    a_s  = h + (size_t)N * OUT_F;
    float*    a_d  = a_s + N;
    float*    ew   = a_d + N;
    float*    ssum = ew + T;
    unsigned* smax = (unsigned*)(ssum + N);

    gat_init<<<(N * OUT_F + 255) / 256, 256, 0, stream>>>(out, bias, smax, ssum, N);
    gat_gemm<<<(N + 127) / 128, 256, 0, stream>>>(x, W, att_src, att_dst, h, a_s, a_d, N);
    gat_logits<<<(T + 255) / 256, 256, 0, stream>>>(ei, a_s, a_d, ew, smax, E, T);
    gat_exp<<<(T + 255) / 256, 256, 0, stream>>>(ei, ew, smax, ssum, E, T);
    gat_aggregate<<<(T + 7) / 8, 256, 0, stream>>>(ei, h, ew, ssum, out, E, T);
}